// GATModel_7705171329594
// MI455X (gfx1250) — compile-verified
//
#include <hip/hip_runtime.h>

#define NN 20000
#define FMV 128
#define HH 8
#define OCV 128
#define EEDGE 160000
#define ETOT (EEDGE + NN)
#define CIRCN 504
#define MIRN (NN - CIRCN)
#define NEG_SLOPE 0.2f
#define HFM (HH * FMV)    // 1024
#define TPAD 132          // LDS row pitch (floats) to dodge bank conflicts

typedef float v2f __attribute__((ext_vector_type(2)));
typedef float v8f __attribute__((ext_vector_type(8)));

// ---- ordered-uint encoding for float atomicMax (handles negatives) ----
static __device__ __forceinline__ unsigned f2ord(float f) {
  unsigned u = __float_as_uint(f);
  return u ^ ((unsigned)((int)u >> 31) | 0x80000000u);
}
static __device__ __forceinline__ float ord2f(unsigned u) {
  unsigned v = (u & 0x80000000u) ? (u ^ 0x80000000u) : ~u;
  return __uint_as_float(v);
}
#define ORD_NEG_INF 0x007FFFFFu  // f2ord(-inf)

// ---- async stage of a contiguous 16x128 A tile (row stride 128) into LDS (pitch TPAD) ----
// Issues 2 x b128 async LDS loads per thread (256 threads: 512 quads = 2048 floats).
static __device__ __forceinline__ void stage_issue(const float* __restrict__ Asrc,
                                                   float* __restrict__ tile) {
  const int t = threadIdx.x;  // 0..255
  #pragma unroll
  for (int it = 0; it < 2; ++it) {
    const int q   = t + it * 256;        // quad id 0..511
    const int row = q >> 5;              // 0..15
    const int c4  = (q & 31) * 4;        // 0..124
    const unsigned lds = (unsigned)(unsigned long long)(tile + row * TPAD + c4);
    const unsigned long long ga = (unsigned long long)(Asrc + row * FMV + c4);
    asm volatile("global_load_async_to_lds_b128 %0, %1, off"
                 :: "v"(lds), "v"(ga) : "memory");
  }
}
static __device__ __forceinline__ void stage_commit() {
  asm volatile("s_wait_asynccnt 0" ::: "memory");
  __syncthreads();
}

// ---- WMMA f32 16x16x4 tile MAC; A fragment from LDS tile (pitch TPAD), B from global.
// B(k,n) = B[k*BRS + (colBase+n)*BCS]
template<int BRS, int BCS>
static __device__ __forceinline__ v8f wmma_mm_lds(const float* __restrict__ tileA,
                                                  const float* __restrict__ B,
                                                  int colBase, int nrem, int K, v8f acc)
{
  const int lane = threadIdx.x & 31;
  const int hf   = lane >> 4;     // K-pair select
  const int lm   = lane & 15;     // row/col within tile
  const float* ap = tileA + lm * TPAD;
  const float* bp = B + (long)(colBase + lm) * BCS;
  const bool bm = lm < nrem;
  for (int k = 0; k < K; k += 4) {
    const int ks = k + 2 * hf;
    v2f a, b;
    a.x = ap[ks];
    a.y = ap[ks + 1];
    if (BRS == 1) {
      if (bm) b = *(const v2f*)(bp + ks);
      else    { b.x = 0.0f; b.y = 0.0f; }
    } else {
      b.x = bm ? bp[(long)ks * BRS]       : 0.0f;
      b.y = bm ? bp[(long)(ks + 1) * BRS] : 0.0f;
    }
    acc = __builtin_amdgcn_wmma_f32_16x16x4_f32(false, a, false, b, (short)0, acc, false, false);
  }
  return acc;
}

static __device__ __forceinline__ void wmma_store(float* __restrict__ C, int rowBase, int ldc, int colBase,
                                                  int mrem, int nrem, v8f acc,
                                                  const float* __restrict__ colBias)
{
  const int lane = threadIdx.x & 31;
  const int hf   = lane >> 4;
  const int lm   = lane & 15;
  if (lm >= nrem) return;
  const float bv = colBias ? colBias[colBase + lm] : 0.0f;
  #pragma unroll
  for (int r = 0; r < 8; ++r) {
    const int m = r + 8 * hf;   // C/D layout: VGPR r -> row r (lanes 0-15) / r+8 (lanes 16-31)
    if (m < mrem)
      C[(long)(rowBase + m) * ldc + colBase + lm] = acc[r] + bv;
  }
}

// ---------------- kernels ----------------

// h = X(20000x128) @ W(128x1024); block = 8 waves sharing one LDS A tile, 8 column tiles
__global__ void __launch_bounds__(256) k_gemm_xw(const float* __restrict__ X,
                                                 const float* __restrict__ W,
                                                 float* __restrict__ Hout)
{
  __shared__ float tileA[16 * TPAD];
  const int mt    = blockIdx.x >> 3;        // 0..1249
  const int nbase = (blockIdx.x & 7) * 8;   // column-tile base (64 total)
  stage_issue(X + (long)mt * 16 * FMV, tileA);
  stage_commit();
  const int nt = nbase + (threadIdx.x >> 5);   // always < 64
  v8f acc = {};
  acc = wmma_mm_lds<HFM, 1>(tileA, W, nt * 16, 16, FMV, acc);
  wmma_store(Hout, mt * 16, HFM, nt * 16, 16, 16, acc, nullptr);
}

// per-(node,head) dots: s = <h[n,h,:], as[h,:]>, d = <h[n,h,:], ad[h,:]>
__global__ void k_sd(const float* __restrict__ Hbuf, const float* __restrict__ As,
                     const float* __restrict__ Ad, float* __restrict__ S, float* __restrict__ D)
{
  const int wave = blockIdx.x * (blockDim.x >> 5) + (threadIdx.x >> 5);
  if (wave >= NN * HH) return;
  const int n = wave / HH, h = wave % HH;
  const int lane = threadIdx.x & 31;
  const float* hp = Hbuf + (long)n * HFM + h * FMV;
  const float* as = As + h * FMV;
  const float* ad = Ad + h * FMV;
  float ss = 0.0f, dd = 0.0f;
  #pragma unroll
  for (int f = lane; f < FMV; f += 32) { const float v = hp[f]; ss += v * as[f]; dd += v * ad[f]; }
  #pragma unroll
  for (int o = 16; o > 0; o >>= 1) { ss += __shfl_down(ss, o, 32); dd += __shfl_down(dd, o, 32); }
  if (lane == 0) { S[wave] = ss; D[wave] = dd; }
}

// zero accumulator, init segment-max to -inf, zero denominators
__global__ void k_init_layer(float* __restrict__ Xacc, unsigned* __restrict__ M, float* __restrict__ Den)
{
  const long total = (long)NN * FMV;
  for (long i = blockIdx.x * (long)blockDim.x + threadIdx.x; i < total; i += (long)gridDim.x * blockDim.x) {
    Xacc[i] = 0.0f;
    if (i < (long)NN * HH) { M[i] = ORD_NEG_INF; Den[i] = 0.0f; }
  }
}

// edge logits (leaky relu) + segment max over dst
__global__ void k_logits(const float* __restrict__ S, const float* __restrict__ D,
                         const int* __restrict__ esrc, const int* __restrict__ edst,
                         float* __restrict__ Ebuf, unsigned* __restrict__ M)
{
  const long total = (long)ETOT * HH;
  for (long i = blockIdx.x * (long)blockDim.x + threadIdx.x; i < total; i += (long)gridDim.x * blockDim.x) {
    const int e = (int)(i >> 3), h = (int)(i & 7);
    const int s = (e < EEDGE) ? esrc[e] : (e - EEDGE);
    const int d = (e < EEDGE) ? edst[e] : (e - EEDGE);
    float v = S[s * HH + h] + D[d * HH + h];
    v = (v > 0.0f) ? v : NEG_SLOPE * v;
    Ebuf[i] = v;
    atomicMax(&M[d * HH + h], f2ord(v));
  }
}

// exp(e - max) + segment sum of denominators
__global__ void k_expsum(const int* __restrict__ esrc, const int* __restrict__ edst,
                         float* __restrict__ Ebuf, const unsigned* __restrict__ M,
                         float* __restrict__ Den)
{
  const long total = (long)ETOT * HH;
  for (long i = blockIdx.x * (long)blockDim.x + threadIdx.x; i < total; i += (long)gridDim.x * blockDim.x) {
    const int e = (int)(i >> 3), h = (int)(i & 7);
    const int d = (e < EEDGE) ? edst[e] : (e - EEDGE);
    const float v = __expf(Ebuf[i] - ord2f(M[d * HH + h]));
    Ebuf[i] = v;
    atomicAdd(&Den[d * HH + h], v);
  }
}

// one wave per edge: Xacc[dst,f] += sum_h alpha[h]/H * h[src,h,f]; float4 reads, 128 atomics/edge
__global__ void k_aggregate(const float* __restrict__ Hbuf, const float* __restrict__ Ebuf,
                            const float* __restrict__ Den,
                            const int* __restrict__ esrc, const int* __restrict__ edst,
                            float* __restrict__ Xacc)
{
  const int e = blockIdx.x * (blockDim.x >> 5) + (threadIdx.x >> 5);
  if (e >= ETOT) return;
  const int lane = threadIdx.x & 31;
  const int s = (e < EEDGE) ? esrc[e] : (e - EEDGE);
  const int d = (e < EEDGE) ? edst[e] : (e - EEDGE);
  float av = 0.0f;
  if (lane < HH)
    av = 0.125f * Ebuf[(long)e * HH + lane] / (Den[d * HH + lane] + 1e-16f);
  float alpha[HH];
  #pragma unroll
  for (int h = 0; h < HH; ++h) alpha[h] = __shfl(av, h, 32);
  const float4* hp = (const float4*)(Hbuf + (long)s * HFM);
  float4 acc = make_float4(0.f, 0.f, 0.f, 0.f);
  #pragma unroll
  for (int h = 0; h < HH; ++h) {
    const float4 v = hp[h * (FMV / 4) + lane];
    acc.x += v.x * alpha[h]; acc.y += v.y * alpha[h];
    acc.z += v.z * alpha[h]; acc.w += v.w * alpha[h];
  }
  float* out = Xacc + (long)d * FMV + lane * 4;
  atomicAdd(out + 0, acc.x); atomicAdd(out + 1, acc.y);
  atomicAdd(out + 2, acc.z); atomicAdd(out + 3, acc.w);
}

// x = relu(acc + b)
__global__ void k_finalize(float* __restrict__ Xacc, const float* __restrict__ b)
{
  const long total = (long)NN * FMV;
  for (long i = blockIdx.x * (long)blockDim.x + threadIdx.x; i < total; i += (long)gridDim.x * blockDim.x)
    Xacc[i] = fmaxf(Xacc[i] + b[i & (FMV - 1)], 0.0f);
}

// emb = x1 @ Wc[:,0,:]^T + x2 @ Wc[:,1,:]^T + bc   (N x OC); 8 waves = 8 column tiles
__global__ void __launch_bounds__(256) k_gemm_emb(const float* __restrict__ X1,
                                                  const float* __restrict__ X2,
                                                  const float* __restrict__ Wc,
                                                  const float* __restrict__ bc,
                                                  float* __restrict__ Emb)
{
  __shared__ float tA1[16 * TPAD];
  __shared__ float tA2[16 * TPAD];
  const int mt = blockIdx.x;   // 0..1249
  stage_issue(X1 + (long)mt * 16 * FMV, tA1);
  stage_issue(X2 + (long)mt * 16 * FMV, tA2);
  stage_commit();
  const int nt = threadIdx.x >> 5;   // 0..7 (OC/16 == 8)
  v8f acc = {};
  // B(k,o) = Wc[o*2*FM + c*FM + k] -> BRS = 1, BCS = 2*FM
  acc = wmma_mm_lds<1, 2 * FMV>(tA1, Wc,       nt * 16, 16, FMV, acc);
  acc = wmma_mm_lds<1, 2 * FMV>(tA2, Wc + FMV, nt * 16, 16, FMV, acc);
  wmma_store(Emb, mt * 16, OCV, nt * 16, 16, 16, acc, bc);
}

// Out = circ @ mir^T (504 x 19496); one LDS A tile per block reused by 8 column tiles
#define CM_MT ((CIRCN + 15) / 16)          // 32
#define CM_NT ((MIRN + 15) / 16)           // 1219
#define CM_NTG ((CM_NT + 7) / 8)           // 153
__global__ void __launch_bounds__(256) k_gemm_cm(const float* __restrict__ Emb,
                                                 float* __restrict__ Out)
{
  __shared__ float tileA[16 * TPAD];
  const int mt  = blockIdx.x / CM_NTG;        // 0..31
  const int ntb = (blockIdx.x % CM_NTG) * 8;
  stage_issue(Emb + (long)mt * 16 * OCV, tileA);   // rows >=504 staged but never stored
  stage_commit();
  const int nt = ntb + (threadIdx.x >> 5);
  if (nt < CM_NT) {
    const int mrem = min(16, CIRCN - mt * 16);
    const int nrem = min(16, MIRN - nt * 16);
    v8f acc = {};
    // B(k,j) = Emb[(CIRC+j)*OC + k] -> BRS = 1, BCS = OC
    acc = wmma_mm_lds<1, OCV>(tileA, Emb + (long)CIRCN * OCV, nt * 16, nrem, OCV, acc);
    wmma_store(Out, mt * 16, MIRN, nt * 16, mrem, nrem, acc, nullptr);
  }
}

// copy circ / mir blocks into concatenated output
__global__ void k_copy_out(const float* __restrict__ Emb, float* __restrict__ Out)
{
  const long OFF_C = (long)CIRCN * MIRN;
  const long OFF_M = OFF_C + (long)CIRCN * OCV;
  const long total = (long)NN * OCV;
  const long split = (long)CIRCN * OCV;
  for (long i = blockIdx.x * (long)blockDim.x + threadIdx.x; i < total; i += (long)gridDim.x * blockDim.x) {
    if (i < split) Out[OFF_C + i] = Emb[i];
    else           Out[OFF_M + (i - split)] = Emb[i];
  }
}

extern "C" void kernel_launch(void* const* d_in, const int* in_sizes, int n_in,
                              void* d_out, int out_size, void* d_ws, size_t ws_size,
                              hipStream_t stream) {
  const float* x   = (const float*)d_in[0];
  const int*   ei  = (const int*)d_in[1];
  const float* W1  = (const float*)d_in[2];
  const float* as1 = (const float*)d_in[3];
  const float* ad1 = (const float*)d_in[4];
  const float* b1  = (const float*)d_in[5];
  const float* W2  = (const float*)d_in[6];
  const float* as2 = (const float*)d_in[7];
  const float* ad2 = (const float*)d_in[8];
  const float* b2  = (const float*)d_in[9];
  const float* Wc  = (const float*)d_in[10];
  const float* bc  = (const float*)d_in[11];
  float* out = (float*)d_out;

  const int* esrc = ei;
  const int* edst = ei + EEDGE;

  // workspace carve-up (floats)
  float* ws   = (float*)d_ws;
  float* hbuf = ws;                         // NN*1024
  float* x1   = hbuf + (long)NN * HFM;      // NN*128
  float* x2   = x1 + (long)NN * FMV;        // NN*128
  float* emb  = x2 + (long)NN * FMV;        // NN*128
  float* sbuf = emb + (long)NN * OCV;       // NN*8
  float* dbuf = sbuf + (long)NN * HH;       // NN*8
  unsigned* mbuf = (unsigned*)(dbuf + (long)NN * HH);  // NN*8
  float* den  = (float*)(mbuf + (long)NN * HH);        // NN*8
  float* ebuf = den + (long)NN * HH;                   // ETOT*8

  const int T = 256;
  const dim3 gGemmXW((NN / 16) * 8, 1, 1);       // 10000 blocks, 8 col-tiles each
  const dim3 gSD((NN * HH + 7) / 8, 1, 1);       // 20000
  const dim3 gEl(5625, 1, 1);                    // grid-stride over ETOT*8
  const dim3 gAgg((ETOT + 7) / 8, 1, 1);         // 22500
  const dim3 gGS(10000, 1, 1);                   // grid-stride 2.56M
  const dim3 gEmb(NN / 16, 1, 1);                // 1250
  const dim3 gCM(CM_MT * CM_NTG, 1, 1);          // 32*153 = 4896

  // ---- layer 1 ----
  k_init_layer<<<gGS, T, 0, stream>>>(x1, mbuf, den);
  k_gemm_xw<<<gGemmXW, T, 0, stream>>>(x, W1, hbuf);
  k_sd<<<gSD, T, 0, stream>>>(hbuf, as1, ad1, sbuf, dbuf);
  k_logits<<<gEl, T, 0, stream>>>(sbuf, dbuf, esrc, edst, ebuf, mbuf);
  k_expsum<<<gEl, T, 0, stream>>>(esrc, edst, ebuf, mbuf, den);
  k_aggregate<<<gAgg, T, 0, stream>>>(hbuf, ebuf, den, esrc, edst, x1);
  k_finalize<<<gGS, T, 0, stream>>>(x1, b1);

  // ---- layer 2 ----
  k_init_layer<<<gGS, T, 0, stream>>>(x2, mbuf, den);
  k_gemm_xw<<<gGemmXW, T, 0, stream>>>(x1, W2, hbuf);
  k_sd<<<gSD, T, 0, stream>>>(hbuf, as2, ad2, sbuf, dbuf);
  k_logits<<<gEl, T, 0, stream>>>(sbuf, dbuf, esrc, edst, ebuf, mbuf);
  k_expsum<<<gEl, T, 0, stream>>>(esrc, edst, ebuf, mbuf, den);
  k_aggregate<<<gAgg, T, 0, stream>>>(hbuf, ebuf, den, esrc, edst, x2);
  k_finalize<<<gGS, T, 0, stream>>>(x2, b2);

  // ---- classifier + similarity ----
  k_gemm_emb<<<gEmb, T, 0, stream>>>(x1, x2, Wc, bc, emb);
  k_gemm_cm<<<gCM, T, 0, stream>>>(emb, out);
  k_copy_out<<<gGS, T, 0, stream>>>(emb, out);

  (void)in_sizes; (void)n_in; (void)out_size; (void)ws_size;
}